// DecomposedAttention_5738076307479
// MI455X (gfx1250) — compile-verified
//
#include <hip/hip_runtime.h>

// Problem constants (B=2, H=8 -> BH=16 heads)
#define BH    16
#define NSEQ  2048
#define DHEAD 64
#define INV_SCALE (1.0f / 64.0f)

#define CHUNKS 16          // split-K chunks over NSEQ for covariance reduction
#define ROWS_PER_CHUNK (NSEQ / CHUNKS)   // 128
#define TILE_R 32          // rows staged in LDS per iteration

#define QPAD 68            // LDS row stride (64 + 4) -> conflict-free b64 reads

typedef float v2f __attribute__((ext_vector_type(2)));
typedef float v8f __attribute__((ext_vector_type(8)));

// ---------------------------------------------------------------------------
// Kernel 1: per-(head, chunk) partial S = Q^T Q and Mraw = K^T V over a chunk
// of 128 rows. grid = BH*CHUNKS blocks, 256 threads. Fixed-order accumulation.
// ---------------------------------------------------------------------------
__global__ void __launch_bounds__(256)
cov_kernel(const float* __restrict__ Q, const float* __restrict__ K,
           const float* __restrict__ V, float* __restrict__ pS,
           float* __restrict__ pM) {
    const int head  = blockIdx.x >> 4;   // /CHUNKS
    const int chunk = blockIdx.x & (CHUNKS - 1);

    __shared__ float Qs[TILE_R][DHEAD];
    __shared__ float Ks[TILE_R][DHEAD];
    __shared__ float Vs[TILE_R][DHEAD];

    const int t  = threadIdx.x;
    const int i  = t >> 2;               // output row 0..63
    const int j0 = (t & 3) << 4;         // output col block of 16

    const float* Qh = Q + (size_t)head * NSEQ * DHEAD;
    const float* Kh = K + (size_t)head * NSEQ * DHEAD;
    const float* Vh = V + (size_t)head * NSEQ * DHEAD;

    float sacc[16];
    float macc[16];
#pragma unroll
    for (int jj = 0; jj < 16; ++jj) { sacc[jj] = 0.0f; macc[jj] = 0.0f; }

    const int row0 = chunk * ROWS_PER_CHUNK;
    for (int base = row0; base < row0 + ROWS_PER_CHUNK; base += TILE_R) {
        // cooperative coalesced load of 32x64 tiles of Q, K, V
#pragma unroll
        for (int e = 0; e < (TILE_R * DHEAD) / 256; ++e) {
            const int lin = e * 256 + t;
            const int r = lin >> 6;
            const int c = lin & 63;
            const size_t g = (size_t)(base + r) * DHEAD + c;
            Qs[r][c] = Qh[g];
            Ks[r][c] = Kh[g];
            Vs[r][c] = Vh[g];
        }
        __syncthreads();

        for (int r = 0; r < TILE_R; ++r) {
            const float qi = Qs[r][i];
            const float ki = Ks[r][i];
#pragma unroll
            for (int jj = 0; jj < 16; ++jj) {
                sacc[jj] = fmaf(qi, Qs[r][j0 + jj], sacc[jj]);
                macc[jj] = fmaf(ki, Vs[r][j0 + jj], macc[jj]);
            }
        }
        __syncthreads();
    }

    float* pSh = pS + ((size_t)head * CHUNKS + chunk) * DHEAD * DHEAD;
    float* pMh = pM + ((size_t)head * CHUNKS + chunk) * DHEAD * DHEAD;
#pragma unroll
    for (int jj = 0; jj < 16; ++jj) {
        pSh[i * DHEAD + j0 + jj] = sacc[jj];
        pMh[i * DHEAD + j0 + jj] = macc[jj];
    }
}

// ---------------------------------------------------------------------------
// Kernel 2: per head, reduce partials -> S, M (M scaled by 1/64), then
// P = S @ M (64x64x64). grid = BH blocks, 256 threads.
// ---------------------------------------------------------------------------
__global__ void __launch_bounds__(256)
reduce_p_kernel(const float* __restrict__ pS, const float* __restrict__ pM,
                float* __restrict__ P) {
    const int head = blockIdx.x;
    __shared__ float Ssh[DHEAD][DHEAD];
    __shared__ float Msh[DHEAD][DHEAD];

    const int t  = threadIdx.x;
    const int i  = t >> 2;
    const int j0 = (t & 3) << 4;

#pragma unroll
    for (int jj = 0; jj < 16; ++jj) {
        float s = 0.0f, m = 0.0f;
        for (int c = 0; c < CHUNKS; ++c) {
            const size_t off = ((size_t)head * CHUNKS + c) * DHEAD * DHEAD
                             + (size_t)i * DHEAD + j0 + jj;
            s += pS[off];
            m += pM[off];
        }
        Ssh[i][j0 + jj] = s;
        Msh[i][j0 + jj] = m * INV_SCALE;
    }
    __syncthreads();

    float* Ph = P + (size_t)head * DHEAD * DHEAD;
#pragma unroll
    for (int jj = 0; jj < 16; ++jj) {
        float acc = 0.0f;
        for (int k = 0; k < DHEAD; ++k)
            acc = fmaf(Ssh[i][k], Msh[k][j0 + jj], acc);
        Ph[i * DHEAD + j0 + jj] = acc;
    }
}

// ---------------------------------------------------------------------------
// Kernel 3: T = Q @ P  ([NSEQ x 64] per head). grid = BH * (NSEQ/256) blocks,
// 256 threads, one row per thread; P cached in LDS, Q row in registers.
// ---------------------------------------------------------------------------
__global__ void __launch_bounds__(256)
qp_kernel(const float* __restrict__ Q, const float* __restrict__ P,
          float* __restrict__ T) {
    const int head = blockIdx.x >> 3;            // NSEQ/256 == 8 blocks/head
    const int rowb = (blockIdx.x & 7) * 256;
    __shared__ float Psh[DHEAD][DHEAD];

    const int t = threadIdx.x;
    const float* Ph = P + (size_t)head * DHEAD * DHEAD;
#pragma unroll
    for (int e = 0; e < (DHEAD * DHEAD) / 256; ++e) {
        const int lin = e * 256 + t;
        Psh[lin >> 6][lin & 63] = Ph[lin];
    }
    __syncthreads();

    const int row = rowb + t;
    const float* q = Q + ((size_t)head * NSEQ + row) * DHEAD;

    float qr[DHEAD];
#pragma unroll
    for (int k4 = 0; k4 < DHEAD / 4; ++k4) {
        const float4 v = ((const float4*)q)[k4];
        qr[4 * k4 + 0] = v.x;
        qr[4 * k4 + 1] = v.y;
        qr[4 * k4 + 2] = v.z;
        qr[4 * k4 + 3] = v.w;
    }

    float* Trow = T + ((size_t)head * NSEQ + row) * DHEAD;
    for (int d = 0; d < DHEAD; ++d) {
        float acc = 0.0f;
#pragma unroll
        for (int k = 0; k < DHEAD; ++k)
            acc = fmaf(qr[k], Psh[k][d], acc);   // Psh broadcast across wave
        Trow[d] = acc;
    }
}

// ---------------------------------------------------------------------------
// Kernel 4: Out = T @ Q^T, fp32 WMMA, block-tiled for L2-traffic reduction.
//
// Block (256 thr = 8 waves) computes a 128x128 output region:
//   - wave w owns output rows [rb + 16w, rb + 16w + 16)
//   - A fragments (16 rows of T, full K=64) preloaded into 32 VGPRs per lane
//   - 128x64 Q tile staged in LDS (stride QPAD=68 -> conflict-free b64 reads)
//   - wave sweeps 8 column tiles: 16 chained v_wmma_f32_16x16x4_f32 each
// Per block: 64KB read + 64KB written (1x amplification) vs 8x before.
// ---------------------------------------------------------------------------
__global__ void __launch_bounds__(256)
tqt_kernel(const float* __restrict__ T, const float* __restrict__ Q,
           float* __restrict__ Out) {
    const int head = blockIdx.x >> 8;            // 256 regions per head
    const int reg  = blockIdx.x & 255;
    const int rb   = (reg >> 4) << 7;            // region row base (x128)
    const int cb   = (reg & 15) << 7;            // region col base (x128)

    const int t    = threadIdx.x;
    const int wid  = t >> 5;
    const int lane = t & 31;
    const int m    = lane & 15;
    const int half = lane >> 4;
    const int koff = half << 1;                  // 0 or 2

    __shared__ float Qs[128][QPAD];

    // ---- stage Q tile (128 rows x 64 cols) into LDS, coalesced float4 ----
    const float* Qg = Q + ((size_t)head * NSEQ + cb) * DHEAD;
#pragma unroll
    for (int e = 0; e < 8; ++e) {
        const int lin = e * 256 + t;             // 0..2047
        const int r   = lin >> 4;                // 0..127
        const int c4  = (lin & 15) << 2;         // 0..60
        const float4 v = *(const float4*)(Qg + (size_t)r * DHEAD + c4);
        *(float4*)&Qs[r][c4] = v;                // (r*68+c4)*4 is 16B-aligned
    }

    // ---- preload this wave's A fragments (16 rows of T, K=64) ----
    const int wrow = rb + (wid << 4);            // wave's output row base
    const float* Trow = T + ((size_t)head * NSEQ + wrow + m) * DHEAD;
    float2 afrag[16];
#pragma unroll
    for (int s = 0; s < 16; ++s)
        afrag[s] = *(const float2*)(Trow + 4 * s + koff);

    __syncthreads();

    float* Oh = Out + (size_t)head * NSEQ * NSEQ;

    // ---- sweep 8 column tiles of the region ----
    for (int tj = 0; tj < 8; ++tj) {
        const int brow = (tj << 4) + m;          // B-frag source row in Qs
        v8f c = {0.f, 0.f, 0.f, 0.f, 0.f, 0.f, 0.f, 0.f};
#pragma unroll
        for (int s = 0; s < 16; ++s) {
            v2f a; a.x = afrag[s].x; a.y = afrag[s].y;
            const float2 b2 = *(const float2*)&Qs[brow][4 * s + koff];
            v2f b; b.x = b2.x; b.y = b2.y;
            // D = A*B + C : emits v_wmma_f32_16x16x4_f32
            c = __builtin_amdgcn_wmma_f32_16x16x4_f32(
                    /*neg_a=*/false, a, /*neg_b=*/false, b,
                    /*c_mod=*/(short)0, c, /*reuse_a=*/false, /*reuse_b=*/false);
        }
        const int col = cb + (tj << 4) + m;
#pragma unroll
        for (int v = 0; v < 8; ++v) {
            const int row = wrow + (half << 3) + v;
            Oh[(size_t)row * NSEQ + col] = c[v];
        }
    }
}

// ---------------------------------------------------------------------------
// Launch: inputs in dict order {X (dead), Q, K, V}; output fp32 [2,8,2048,2048]
// Workspace layout (floats): pS | pM | P | T   (~16.3 MB total)
// ---------------------------------------------------------------------------
extern "C" void kernel_launch(void* const* d_in, const int* in_sizes, int n_in,
                              void* d_out, int out_size, void* d_ws, size_t ws_size,
                              hipStream_t stream) {
    (void)in_sizes; (void)n_in; (void)out_size; (void)ws_size;

    const float* Q = (const float*)d_in[1];
    const float* K = (const float*)d_in[2];
    const float* V = (const float*)d_in[3];
    float* Out = (float*)d_out;

    float* ws = (float*)d_ws;
    const size_t nPart = (size_t)BH * CHUNKS * DHEAD * DHEAD;   // 1,048,576
    float* pS = ws;
    float* pM = pS + nPart;
    float* P  = pM + nPart;                                     // BH*4096
    float* T  = P + (size_t)BH * DHEAD * DHEAD;                 // BH*NSEQ*DHEAD

    // 1) partial covariances S = Q^T Q, Mraw = K^T V (split-K over rows)
    cov_kernel<<<BH * CHUNKS, 256, 0, stream>>>(Q, K, V, pS, pM);

    // 2) reduce + P = S @ (Mraw/64)
    reduce_p_kernel<<<BH, 256, 0, stream>>>(pS, pM, P);

    // 3) T = Q @ P
    qp_kernel<<<BH * (NSEQ / 256), 256, 0, stream>>>(Q, P, T);

    // 4) Out = T @ Q^T  (WMMA fp32, 128x128 region per block, LDS-tiled)
    tqt_kernel<<<BH * 256, 256, 0, stream>>>(T, Q, Out);
}